// HeteroGNN_GraphConv_55327768707100
// MI455X (gfx1250) — compile-verified
//
#include <hip/hip_runtime.h>
#include <hip/hip_bf16.h>

// ---------------------------------------------------------------------------
// HeteroGNN GraphConv for MI455X (gfx1250, wave32).
// Dense GEMMs (K=64) run as V_WMMA_F32_16X16X4_F32 chains in fp32: the
// workload is HBM-bound (~2 GB edge gather/scatter + several 64 MB activation
// passes vs only ~21 GFLOP dense), so fp32 matrix ops keep exact reference
// semantics at zero cost. Weight matrices are staged in LDS (32 KB/WGP block)
// so vector-memory bandwidth is reserved for the activation stream.
// ---------------------------------------------------------------------------

#define N_NODES 262144
#define NEDGE   1000000
#define BGRAPH  128
#define HDIM    64
#define BN_EPS  1e-5f
#define SLOPE   0.01f

typedef __attribute__((ext_vector_type(2))) float v2f;
typedef __attribute__((ext_vector_type(8))) float v8f;

__device__ __forceinline__ v8f wmma4(v2f a, v2f b, v8f c) {
  // V_WMMA_F32_16X16X4_F32: D = A(16x4) * B(4x16) + C(16x16), fp32
  return __builtin_amdgcn_wmma_f32_16x16x4_f32(
      /*neg_a=*/false, a, /*neg_b=*/false, b,
      /*c_mod=*/(short)0, c, /*reuse_a=*/false, /*reuse_b=*/false);
}

// Y[r][n] = leakyrelu( BN( X1@W1 (+ X2@W2) + bias ) )
// X row-major [nrows][64]; W row-major [64][64]. One wave -> 16x64 output tile,
// 8 waves per block. W1 (and W2) staged in LDS, B fragments served by ds_load.
// In-place Y==X2 is safe: each row tile is owned by exactly one wave and all
// X2 reads feed the WMMA accumulators that the stores depend on.
template <bool HAS_X2>
__global__ __launch_bounds__(256) void fused_conv_kernel(
    const float* X1, const float* __restrict__ W1,
    const float* X2, const float* __restrict__ W2,
    const float* __restrict__ bias,
    const float* __restrict__ bng, const float* __restrict__ bnb,
    const float* __restrict__ bnm, const float* __restrict__ bnv,
    float* Y, int nrows)
{
  constexpr int WCNT = HAS_X2 ? 2 : 1;
  __shared__ float ldsW[WCNT * 4096];          // W1 [64][64] (+ W2) row-major

  // Cooperative weight staging: 256 threads x WCNT*4 float4s.
  {
    const int tid = threadIdx.x;
#pragma unroll
    for (int i = 0; i < WCNT * 4; ++i) {
      const int idx = (i * 256 + tid) * 4;     // float index, 0..WCNT*4096-4
      float4 v;
      if (i < 4) v = *(const float4*)(W1 + idx);
      else       v = *(const float4*)(W2 + (idx - 4096));
      *(float4*)(&ldsW[idx]) = v;
    }
  }
  __syncthreads();

  const int wave = threadIdx.x >> 5;
  const int lane = threadIdx.x & 31;
  const int tile = blockIdx.x * 8 + wave;
  const int r0 = tile * 16;
  if (r0 >= nrows) return;                     // wave-uniform: EXEC all-ones

  const int lr   = lane & 15;                  // A: row-in-tile; B/C/D: col
  const int half = lane >> 4;                  // lane half selects K / M block

  v8f acc[4] = {};                             // 4 N-tiles of 16 cols

  const float* x1row = X1 + (size_t)(r0 + lr) * 64;
  const float* x2row = HAS_X2 ? (X2 + (size_t)(r0 + lr) * 64) : nullptr;

#pragma unroll
  for (int s = 0; s < 16; ++s) {               // K = 64 = 16 steps of K=4
    const int kb = 4 * s + 2 * half;           // A layout: vgpr j -> K = kb+j
    v2f a1;
    a1.x = x1row[kb];
    a1.y = x1row[kb + 1];
#pragma unroll
    for (int t = 0; t < 4; ++t) {
      const int n = t * 16 + lr;               // B layout: vgpr j -> K=kb+j
      v2f b1;
      b1.x = ldsW[kb * 64 + n];
      b1.y = ldsW[(kb + 1) * 64 + n];
      acc[t] = wmma4(a1, b1, acc[t]);
    }
    if (HAS_X2) {
      v2f a2;
      a2.x = x2row[kb];
      a2.y = x2row[kb + 1];
#pragma unroll
      for (int t = 0; t < 4; ++t) {
        const int n = t * 16 + lr;
        v2f b2;
        b2.x = ldsW[4096 + kb * 64 + n];
        b2.y = ldsW[4096 + (kb + 1) * 64 + n];
        acc[t] = wmma4(a2, b2, acc[t]);
      }
    }
  }

#pragma unroll
  for (int t = 0; t < 4; ++t) {
    const int n = t * 16 + lr;
    const float bsv   = bias[n];
    const float scale = bng[n] * rsqrtf(bnv[n] + BN_EPS);
    const float shift = bnb[n] - bnm[n] * scale;
#pragma unroll
    for (int j = 0; j < 8; ++j) {
      const int m = j + 8 * half;              // C/D layout: vgpr j -> row m
      float v = (acc[t][j] + bsv) * scale + shift;
      v = (v > 0.0f) ? v : SLOPE * v;
      Y[(size_t)(r0 + m) * 64 + n] = v;
    }
  }
}

__global__ void fill_kernel(float* __restrict__ p, float v, size_t n) {
  size_t i = (size_t)blockIdx.x * blockDim.x + threadIdx.x;
  if (i < n) p[i] = v;
}

// agg[dst] += h[src] over all edges; 16 threads per edge, float4 each.
__global__ void scatter_add_kernel(const float* __restrict__ h,
                                   const int* __restrict__ edges,  // [2][E]
                                   float* __restrict__ agg, int E) {
  long long gid = (long long)blockIdx.x * blockDim.x + threadIdx.x;
  int e = (int)(gid >> 4);
  if (e >= E) return;
  int f4  = ((int)gid & 15) * 4;
  int src = edges[e];
  int dst = edges[E + e];
  const float4 v = *(const float4*)(h + (size_t)src * 64 + f4);
  float* p = agg + (size_t)dst * 64 + f4;
  atomicAdd(p + 0, v.x);
  atomicAdd(p + 1, v.y);
  atomicAdd(p + 2, v.z);
  atomicAdd(p + 3, v.w);
}

__device__ __forceinline__ void atomicMaxFloat(float* addr, float v) {
  if (v >= 0.0f) atomicMax((int*)addr, __float_as_int(v));
  else           atomicMin((unsigned int*)addr, __float_as_uint(v));
}

// Per-graph sum / max / count accumulation. 16 threads per node.
__global__ void pool_kernel(const float* __restrict__ h,
                            const int* __restrict__ batch,
                            float* __restrict__ sum, float* __restrict__ mx,
                            float* __restrict__ cnt, int nrows) {
  long long gid = (long long)blockIdx.x * blockDim.x + threadIdx.x;
  int node = (int)(gid >> 4);
  if (node >= nrows) return;
  int f4 = ((int)gid & 15) * 4;
  int g = batch[node];
  const float4 v = *(const float4*)(h + (size_t)node * 64 + f4);
  float* ps = sum + (size_t)g * 64 + f4;
  float* pm = mx + (size_t)g * 64 + f4;
  atomicAdd(ps + 0, v.x); atomicAdd(ps + 1, v.y);
  atomicAdd(ps + 2, v.z); atomicAdd(ps + 3, v.w);
  atomicMaxFloat(pm + 0, v.x); atomicMaxFloat(pm + 1, v.y);
  atomicMaxFloat(pm + 2, v.z); atomicMaxFloat(pm + 3, v.w);
  if (f4 == 0) atomicAdd(&cnt[g], 1.0f);
}

// rep[g] = [mean_a || max_a || mean_b || max_b]  -> [B, 256]
__global__ void rep_kernel(const float* __restrict__ sum_a, const float* __restrict__ mx_a,
                           const float* __restrict__ cnt_a,
                           const float* __restrict__ sum_b, const float* __restrict__ mx_b,
                           const float* __restrict__ cnt_b,
                           float* __restrict__ rep) {
  int gid = blockIdx.x * blockDim.x + threadIdx.x;
  if (gid >= BGRAPH * 256) return;
  int g = gid >> 8;
  int c = gid & 255;
  int seg = c >> 6;
  int f = c & 63;
  float out;
  if (seg == 0)      out = sum_a[g * 64 + f] / fmaxf(cnt_a[g], 1.0f);
  else if (seg == 1) out = (cnt_a[g] > 0.0f) ? mx_a[g * 64 + f] : 0.0f;
  else if (seg == 2) out = sum_b[g * 64 + f] / fmaxf(cnt_b[g], 1.0f);
  else               out = (cnt_b[g] > 0.0f) ? mx_b[g * 64 + f] : 0.0f;
  rep[gid] = out;
}

// h = rep @ mlp_W + mlp_b  ([128,256]@[256,64]); tiny -> scalar FMA.
__global__ void mlp_kernel(const float* __restrict__ rep, const float* __restrict__ W,
                           const float* __restrict__ b, float* __restrict__ h) {
  int gid = blockIdx.x * blockDim.x + threadIdx.x;
  if (gid >= BGRAPH * HDIM) return;
  int g = gid >> 6, j = gid & 63;
  float acc = b[j];
  const float* r = rep + (size_t)g * 256;
#pragma unroll 8
  for (int c = 0; c < 256; ++c) acc = fmaf(r[c], W[(size_t)c * 64 + j], acc);
  h[gid] = acc;
}

__global__ void lin_kernel(const float* __restrict__ h, const float* __restrict__ W,
                           const float* __restrict__ b, float* __restrict__ out) {
  int g = blockIdx.x * blockDim.x + threadIdx.x;
  if (g >= BGRAPH) return;
  float acc = b[0];
#pragma unroll
  for (int j = 0; j < 64; ++j) acc = fmaf(h[(size_t)g * 64 + j], W[j], acc);
  out[g] = acc;
}

extern "C" void kernel_launch(void* const* d_in, const int* in_sizes, int n_in,
                              void* d_out, int out_size, void* d_ws, size_t ws_size,
                              hipStream_t stream) {
  const float* x_a     = (const float*)d_in[0];
  const float* x_b     = (const float*)d_in[1];
  const int*   edge_ab = (const int*)d_in[2];
  const int*   edge_ba = (const int*)d_in[3];
  const int*   batch_a = (const int*)d_in[4];
  const int*   batch_b = (const int*)d_in[5];
  const float* proj_Wa = (const float*)d_in[6];
  const float* proj_ba = (const float*)d_in[7];
  const float* proj_Wb = (const float*)d_in[8];
  const float* proj_bb = (const float*)d_in[9];
  const float* bn_a_g  = (const float*)d_in[10];
  const float* bn_a_b  = (const float*)d_in[11];
  const float* bn_a_m  = (const float*)d_in[12];
  const float* bn_a_v  = (const float*)d_in[13];
  const float* bn_b_g  = (const float*)d_in[14];
  const float* bn_b_b  = (const float*)d_in[15];
  const float* bn_b_m  = (const float*)d_in[16];
  const float* bn_b_v  = (const float*)d_in[17];
  const float* Wrel_ab[2]  = {(const float*)d_in[18], (const float*)d_in[24]};
  const float* brel_ab[2]  = {(const float*)d_in[19], (const float*)d_in[25]};
  const float* Wroot_ab[2] = {(const float*)d_in[20], (const float*)d_in[26]};
  const float* Wrel_ba[2]  = {(const float*)d_in[21], (const float*)d_in[27]};
  const float* brel_ba[2]  = {(const float*)d_in[22], (const float*)d_in[28]};
  const float* Wroot_ba[2] = {(const float*)d_in[23], (const float*)d_in[29]};
  const float* mlp_W = (const float*)d_in[30];
  const float* mlp_b = (const float*)d_in[31];
  const float* lin_W = (const float*)d_in[32];
  const float* lin_b = (const float*)d_in[33];

  // Workspace layout (floats)
  const size_t NH = (size_t)N_NODES * HDIM;          // 16.78M floats each
  float* ws    = (float*)d_ws;
  float* ha    = ws;
  float* hb    = ha + NH;
  float* agg_a = hb + NH;
  float* agg_b = agg_a + NH;
  float* sum_a = agg_b + NH;
  float* mx_a  = sum_a + (size_t)BGRAPH * HDIM;
  float* cnt_a = mx_a + (size_t)BGRAPH * HDIM;
  float* sum_b = cnt_a + BGRAPH;
  float* mx_b  = sum_b + (size_t)BGRAPH * HDIM;
  float* cnt_b = mx_b + (size_t)BGRAPH * HDIM;
  float* rep   = cnt_b + BGRAPH;
  float* hmid  = rep + (size_t)BGRAPH * 256;

  const int convGrid    = N_NODES / (16 * 8);                      // 2048
  const int scatterGrid = (int)(((size_t)NEDGE * 16 + 255) / 256); // 62500
  const int fillGridNH  = (int)((NH + 255) / 256);
  const int poolGrid    = (int)(((size_t)N_NODES * 16 + 255) / 256);
  const float NEG_BIG = -3.402823466e38f;

  // Input projection + BN + leaky_relu
  fused_conv_kernel<false><<<convGrid, 256, 0, stream>>>(
      x_a, proj_Wa, nullptr, nullptr, proj_ba,
      bn_a_g, bn_a_b, bn_a_m, bn_a_v, ha, N_NODES);
  fused_conv_kernel<false><<<convGrid, 256, 0, stream>>>(
      x_b, proj_Wb, nullptr, nullptr, proj_bb,
      bn_b_g, bn_b_b, bn_b_m, bn_b_v, hb, N_NODES);

  for (int L = 0; L < 2; ++L) {
    fill_kernel<<<fillGridNH, 256, 0, stream>>>(agg_b, 0.0f, NH);
    fill_kernel<<<fillGridNH, 256, 0, stream>>>(agg_a, 0.0f, NH);
    scatter_add_kernel<<<scatterGrid, 256, 0, stream>>>(ha, edge_ab, agg_b, NEDGE);
    scatter_add_kernel<<<scatterGrid, 256, 0, stream>>>(hb, edge_ba, agg_a, NEDGE);
    // nb -> hb (in place, root term = old hb), BN_b + act
    fused_conv_kernel<true><<<convGrid, 256, 0, stream>>>(
        agg_b, Wrel_ab[L], hb, Wroot_ab[L], brel_ab[L],
        bn_b_g, bn_b_b, bn_b_m, bn_b_v, hb, N_NODES);
    // na -> ha (in place, root term = old ha), BN_a + act
    fused_conv_kernel<true><<<convGrid, 256, 0, stream>>>(
        agg_a, Wrel_ba[L], ha, Wroot_ba[L], brel_ba[L],
        bn_a_g, bn_a_b, bn_a_m, bn_a_v, ha, N_NODES);
  }

  // Pooling
  fill_kernel<<<(BGRAPH * HDIM + 255) / 256, 256, 0, stream>>>(sum_a, 0.0f, BGRAPH * HDIM);
  fill_kernel<<<(BGRAPH * HDIM + 255) / 256, 256, 0, stream>>>(mx_a, NEG_BIG, BGRAPH * HDIM);
  fill_kernel<<<1, 256, 0, stream>>>(cnt_a, 0.0f, BGRAPH);
  fill_kernel<<<(BGRAPH * HDIM + 255) / 256, 256, 0, stream>>>(sum_b, 0.0f, BGRAPH * HDIM);
  fill_kernel<<<(BGRAPH * HDIM + 255) / 256, 256, 0, stream>>>(mx_b, NEG_BIG, BGRAPH * HDIM);
  fill_kernel<<<1, 256, 0, stream>>>(cnt_b, 0.0f, BGRAPH);
  pool_kernel<<<poolGrid, 256, 0, stream>>>(ha, batch_a, sum_a, mx_a, cnt_a, N_NODES);
  pool_kernel<<<poolGrid, 256, 0, stream>>>(hb, batch_b, sum_b, mx_b, cnt_b, N_NODES);

  // Readout head
  rep_kernel<<<(BGRAPH * 256 + 255) / 256, 256, 0, stream>>>(
      sum_a, mx_a, cnt_a, sum_b, mx_b, cnt_b, rep);
  mlp_kernel<<<(BGRAPH * HDIM + 255) / 256, 256, 0, stream>>>(rep, mlp_W, mlp_b, hmid);
  lin_kernel<<<1, BGRAPH, 0, stream>>>(hmid, lin_W, lin_b, (float*)d_out);
}